// GCNLinkPredictor_88553635709600
// MI455X (gfx1250) — compile-verified
//
#include <hip/hip_runtime.h>
#include <hip/hip_bf16.h>

// GCN link predictor for MI455X (gfx1250, wave32).
// Pipeline: degree -> dinv -> [x@W (WMMA f32) -> init(self-loop+bias) ->
//           edge scatter-add -> relu?] x2 -> edge decode (dot+sigmoid).
// Working set (~157MB in d_ws); xw/h/z (51.2MB each) are L2-resident (192MB L2),
// so edge gathers + f32 scatter atomics resolve at L2, not HBM.

typedef __attribute__((ext_vector_type(2))) float v2f;
typedef __attribute__((ext_vector_type(8))) float v8f;

#define D_FEAT 128

// ---------------- utility ----------------
__global__ void k_zero(float* __restrict__ p, int n) {
  int i = blockIdx.x * blockDim.x + threadIdx.x;
  if (i < n) p[i] = 0.0f;
}

__global__ void k_degree(const long long* __restrict__ dst, float* __restrict__ deg, int nE) {
  int e = blockIdx.x * blockDim.x + threadIdx.x;
  if (e < nE) atomicAdd(&deg[dst[e]], 1.0f);
}

__global__ void k_dinv(float* __restrict__ d, int nN) {
  int i = blockIdx.x * blockDim.x + threadIdx.x;
  if (i < nN) d[i] = rsqrtf(d[i] + 1.0f);   // deg includes self-loop
}

// ---------------- WMMA GEMM: C[M x 128] = A[M x 128] @ W[128 x 128] ----------------
// Block = 256 threads (8 waves). Each block: 16 rows of A staged in LDS (padded),
// wave w computes the 16x16 tile at columns [16w, 16w+16). K stepped by 4 using
// V_WMMA_F32_16X16X4_F32 (full fp32 precision). W (64KB) stays hot in L1/L2 and
// its loads are coalesced (16 consecutive floats per half-wave).
#define SA_PITCH 132   // 128 + 4 pad: row stride 132 floats -> 16B aligned, no 64-bank conflicts

__global__ void __launch_bounds__(256) k_gemm128(const float* __restrict__ A,
                                                 const float* __restrict__ W,
                                                 float* __restrict__ C, int M) {
  __shared__ float sA[16 * SA_PITCH];
  const int tid  = threadIdx.x;
  const int wave = tid >> 5;
  const int lane = tid & 31;
  const int half = lane >> 4;   // 0: lanes 0-15, 1: lanes 16-31
  const int l16  = lane & 15;
  const int row0 = blockIdx.x * 16;

  // cooperative load: 16 rows x 128 floats = 512 float4, 2 per thread
  for (int i = tid; i < 512; i += 256) {
    int r = i >> 5;       // row 0..15
    int c = i & 31;       // float4 column
    int gr = row0 + r; if (gr >= M) gr = M - 1;   // clamp (keeps EXEC full)
    float4 v = ((const float4*)(A + (size_t)gr * D_FEAT))[c];
    *(float4*)(&sA[r * SA_PITCH + c * 4]) = v;
  }
  __syncthreads();

  v8f acc = {};
  const int col = (wave << 4) + l16;   // global N index for this lane
  #pragma unroll
  for (int k = 0; k < D_FEAT; k += 4) {
    const int ka = k + (half << 1);    // A layout: lanes 0-15 hold K={k,k+1}, 16-31 hold {k+2,k+3}
    v2f a, b;
    a.x = sA[l16 * SA_PITCH + ka];
    a.y = sA[l16 * SA_PITCH + ka + 1];
    b.x = W[(size_t)ka * D_FEAT + col];        // B mirrored: K split across lane halves
    b.y = W[(size_t)(ka + 1) * D_FEAT + col];
    acc = __builtin_amdgcn_wmma_f32_16x16x4_f32(
        /*neg_a=*/false, a, /*neg_b=*/false, b,
        /*c_mod=*/(short)0, acc, /*reuse_a=*/false, /*reuse_b=*/false);
  }

  // D layout: VGPR j -> row (j + 8*half), col = lane%16 within tile
  #pragma unroll
  for (int j = 0; j < 8; ++j) {
    int m = row0 + j + (half << 3);
    if (m < M) C[(size_t)m * D_FEAT + col] = acc[j];
  }
}

// ---------------- agg init: agg[i] = xw[i] * dinv[i]^2 + bias ----------------
__global__ void k_init_agg(const float* __restrict__ xw, const float* __restrict__ dinv,
                           const float* __restrict__ bias, float* __restrict__ agg, int nN) {
  int gid = blockIdx.x * blockDim.x + threadIdx.x;   // nN * 32 threads
  int i = gid >> 5;
  int q = gid & 31;
  if (i >= nN) return;
  float di = dinv[i];
  float s = di * di;
  float4 v = ((const float4*)(xw + (size_t)i * D_FEAT))[q];
  float4 b = ((const float4*)bias)[q];
  float4 r = { v.x * s + b.x, v.y * s + b.y, v.z * s + b.z, v.w * s + b.w };
  ((float4*)(agg + (size_t)i * D_FEAT))[q] = r;
}

// ---------------- edge scatter: agg[dst] += xw[src] * (dinv[src]*dinv[dst]) ----------------
// One wave per edge: lane q moves float4 q of the 128-float row (coalesced 512B row).
__global__ void k_aggregate(const long long* __restrict__ src, const long long* __restrict__ dst,
                            const float* __restrict__ xw, const float* __restrict__ dinv,
                            float* __restrict__ agg, int nE) {
  int gid = blockIdx.x * blockDim.x + threadIdx.x;
  int e = gid >> 5;
  int q = gid & 31;
  if (e >= nE) return;
  long long s = src[e];
  long long d = dst[e];
  float norm = dinv[s] * dinv[d];
  float4 m = ((const float4*)(xw + (size_t)s * D_FEAT))[q];
  float* a = agg + (size_t)d * D_FEAT + q * 4;
  atomicAdd(a + 0, m.x * norm);
  atomicAdd(a + 1, m.y * norm);
  atomicAdd(a + 2, m.z * norm);
  atomicAdd(a + 3, m.w * norm);
}

__global__ void k_relu(float* __restrict__ p, int n) {
  int i = blockIdx.x * blockDim.x + threadIdx.x;
  if (i < n) p[i] = fmaxf(p[i], 0.0f);
}

// ---------------- decode: sigmoid(z[src].fcw_lo + z[dst].fcw_hi + fcb) ----------------
__global__ void k_decode(const long long* __restrict__ src, const long long* __restrict__ dst,
                         const float* __restrict__ z, const float* __restrict__ fcw,
                         const float* __restrict__ fcb, float* __restrict__ out, int nE) {
  int gid = blockIdx.x * blockDim.x + threadIdx.x;
  int e = gid >> 5;
  int q = gid & 31;
  if (e >= nE) return;
  long long s = src[e];
  long long d = dst[e];
  float4 zs = ((const float4*)(z + (size_t)s * D_FEAT))[q];
  float4 zd = ((const float4*)(z + (size_t)d * D_FEAT))[q];
  float4 ws = ((const float4*)fcw)[q];
  float4 wd = ((const float4*)(fcw + D_FEAT))[q];
  float sum = zs.x * ws.x + zs.y * ws.y + zs.z * ws.z + zs.w * ws.w
            + zd.x * wd.x + zd.y * wd.y + zd.z * wd.z + zd.w * wd.w;
  // wave32 reduction
  #pragma unroll
  for (int m = 16; m >= 1; m >>= 1) sum += __shfl_xor(sum, m, 32);
  if (q == 0) out[e] = 1.0f / (1.0f + expf(-(sum + fcb[0])));
}

// ---------------- launcher ----------------
extern "C" void kernel_launch(void* const* d_in, const int* in_sizes, int n_in,
                              void* d_out, int out_size, void* d_ws, size_t ws_size,
                              hipStream_t stream) {
  const float*      x   = (const float*)d_in[0];
  const long long*  ei  = (const long long*)d_in[1];   // (2, nE) int64
  const float*      w1  = (const float*)d_in[2];
  const float*      b1  = (const float*)d_in[3];
  const float*      w2  = (const float*)d_in[4];
  const float*      b2  = (const float*)d_in[5];
  const float*      fcw = (const float*)d_in[6];
  const float*      fcb = (const float*)d_in[7];
  float*            out = (float*)d_out;

  const int nN = in_sizes[0] / D_FEAT;   // 100000
  const int nE = in_sizes[1] / 2;        // 500000
  const long long* src = ei;
  const long long* dst = ei + nE;

  // workspace layout
  char* ws = (char*)d_ws;
  const size_t FEAT_BYTES = (size_t)nN * D_FEAT * sizeof(float);   // 51.2 MB
  float* dinv = (float*)ws;                                        // nN floats
  float* xw   = (float*)(ws + (size_t)(1 << 20));                  // GEMM output (both layers)
  float* h    = (float*)(ws + (size_t)(1 << 20) + FEAT_BYTES);     // layer-1 agg / hidden
  float* z    = (float*)(ws + (size_t)(1 << 20) + 2 * FEAT_BYTES); // layer-2 agg / embedding

  const int BT = 256;
  dim3 blk(BT);
  int gN    = (nN + BT - 1) / BT;
  int gE    = (nE + BT - 1) / BT;
  int gN32  = ((size_t)nN * 32 + BT - 1) / BT;
  int gE32  = ((size_t)nE * 32 + BT - 1) / BT;
  int gF    = ((size_t)nN * D_FEAT + BT - 1) / BT;
  int gGemm = (nN + 15) / 16;

  // degrees -> dinv
  k_zero<<<gN, blk, 0, stream>>>(dinv, nN);
  k_degree<<<gE, blk, 0, stream>>>(dst, dinv, nE);
  k_dinv<<<gN, blk, 0, stream>>>(dinv, nN);

  // ---- layer 1: h = relu(gcn(x, w1, b1)) ----
  k_gemm128<<<gGemm, blk, 0, stream>>>(x, w1, xw, nN);
  k_init_agg<<<gN32, blk, 0, stream>>>(xw, dinv, b1, h, nN);
  k_aggregate<<<gE32, blk, 0, stream>>>(src, dst, xw, dinv, h, nE);
  k_relu<<<gF, blk, 0, stream>>>(h, nN * D_FEAT);

  // ---- layer 2: z = gcn(h, w2, b2) ----
  k_gemm128<<<gGemm, blk, 0, stream>>>(h, w2, xw, nN);
  k_init_agg<<<gN32, blk, 0, stream>>>(xw, dinv, b2, z, nN);
  k_aggregate<<<gE32, blk, 0, stream>>>(src, dst, xw, dinv, z, nE);

  // ---- decode ----
  k_decode<<<gE32, blk, 0, stream>>>(src, dst, z, fcw, fcb, out, nE);
}